// Net_19782619365547
// MI455X (gfx1250) — compile-verified
//
#include <hip/hip_runtime.h>
#include <cmath>

#define IN_  10
#define H_   100
#define HP   112      // H padded to 7 WMMA n-tiles
#define OUT_ 10
#define L_   5
#define B_   64
#define T_   2048
#define M_   (B_ * T_)   // 131072 rows for the parallel GEMMs

typedef __attribute__((ext_vector_type(2))) float v2f;
typedef __attribute__((ext_vector_type(8))) float v8f;

// ---------------------------------------------------------------------------
// Layer 0 input contribution: xp[m,h] = sum_d x[m,d] * Wih0[h,d] + bih[h] + bhh[h]
// K = 10 -> scalar is the right tool (latency-bound workload, no WMMA win here).
// ---------------------------------------------------------------------------
__global__ void xp0_kernel(const float* __restrict__ x, const float* __restrict__ Wih0,
                           const float* __restrict__ bih, const float* __restrict__ bhh,
                           float* __restrict__ xp) {
    int idx = blockIdx.x * blockDim.x + threadIdx.x;  // over M_*H_
    if (idx >= M_ * H_) return;
    int h = idx % H_;
    int m = idx / H_;
    const float* xr = x + (size_t)m * IN_;
    const float* wr = Wih0 + (size_t)h * IN_;
    float acc = bih[h] + bhh[h];
#pragma unroll
    for (int d = 0; d < IN_; ++d) acc = fmaf(xr[d], wr[d], acc);
    xp[idx] = acc;
}

// ---------------------------------------------------------------------------
// Parallel input-contribution GEMM for layers 1..4:
//   xp[m,n] = sum_k prev[m,k] * Wih[n,k] + bih[n] + bhh[n]
// M = 131072, N = 100 (pad 112), K = 100. fp32 WMMA 16x16x4.
// Block = 224 threads = 7 waves; wave w owns n-tile w; one 16-row strip/block.
// ---------------------------------------------------------------------------
__global__ void __launch_bounds__(224)
xp_gemm_kernel(const float* __restrict__ prev, const float* __restrict__ Wih,
               const float* __restrict__ bih, const float* __restrict__ bhh,
               float* __restrict__ xp) {
    __shared__ float sW[H_][HP];  // sW[k][n] = Wih[n][k]  (B operand, K x N)
    int tid = threadIdx.x;
    for (int i = tid; i < H_ * HP; i += 224) {
        int k = i / HP, n = i % HP;
        sW[k][n] = (n < H_) ? Wih[n * H_ + k] : 0.0f;
    }
    __syncthreads();

    int wave = tid >> 5;           // 0..6 -> n-tile
    int lane = tid & 31;
    int lm   = lane & 15;
    int koff = (lane >> 4) * 2;    // half-wave K split for 16x4 A / 4x16 B frags
    int n0   = wave * 16;

    const float* arow = prev + (size_t)(blockIdx.x * 16 + lm) * H_;
    v8f c = {};
    for (int k = 0; k < H_; k += 4) {
        v2f a, b;
        a.x = arow[k + koff];
        a.y = arow[k + koff + 1];
        b.x = sW[k + koff][n0 + lm];
        b.y = sW[k + koff + 1][n0 + lm];
        c = __builtin_amdgcn_wmma_f32_16x16x4_f32(false, a, false, b, (short)0, c,
                                                  false, false);
    }

    int ncol  = n0 + lm;
    int mbase = blockIdx.x * 16 + (lane >> 4) * 8;  // C layout: M = r + 8*(lane>=16)
    if (ncol < H_) {
        float bias = bih[ncol] + bhh[ncol];
#pragma unroll
        for (int r = 0; r < 8; ++r)
            xp[(size_t)(mbase + r) * H_ + ncol] = c[r] + bias;
    }
}

// ---------------------------------------------------------------------------
// Persistent recurrent scan for one layer (single workgroup, LDS-resident state):
//   h = tanh(xp[:,t,:] + h @ Whh^T),  t = 0..T-1
// h: 64x112 in LDS, Whh^T: 100x112 in LDS (~72 KB total, <<320 KB/WGP).
// 8 waves x up to 4 tiles cover the 4x7 = 28 (16x16) output tiles per step.
// ---------------------------------------------------------------------------
__global__ void __launch_bounds__(256)
scan_kernel(const float* __restrict__ xp, float* __restrict__ out,
            const float* __restrict__ h0, const float* __restrict__ Whh,
            float* __restrict__ hT) {
    __shared__ float sW[H_][HP];  // sW[k][n] = Whh[n][k]
    __shared__ float sH[B_][HP];  // current hidden state, zero-padded cols

    int tid = threadIdx.x;
    for (int i = tid; i < H_ * HP; i += 256) {
        int k = i / HP, n = i % HP;
        sW[k][n] = (n < H_) ? Whh[n * H_ + k] : 0.0f;
    }
    for (int i = tid; i < B_ * HP; i += 256) {
        int m = i / HP, n = i % HP;
        sH[m][n] = (n < H_) ? h0[m * H_ + n] : 0.0f;
    }
    __syncthreads();

    int wave = tid >> 5;          // 0..7
    int lane = tid & 31;
    int lm   = lane & 15;
    int koff = (lane >> 4) * 2;

    for (int t = 0; t < T_; ++t) {
        if (t + 1 < T_)  // warm L2/WGP$ for next step's xp rows
            __builtin_prefetch(&xp[((size_t)(tid & 63) * T_ + (t + 1)) * H_], 0, 0);

        v8f acc[4];
#pragma unroll
        for (int j = 0; j < 4; ++j) {
            int tile = wave + j * 8;
            if (tile < 28) {  // wave-uniform: EXEC stays all-ones for WMMA
                int mt = tile / 7, nt = tile % 7;
                const float* arow = &sH[mt * 16 + lm][0];
                v8f c = {};
                for (int k = 0; k < H_; k += 4) {
                    v2f a, b;
                    a.x = arow[k + koff];
                    a.y = arow[k + koff + 1];
                    b.x = sW[k + koff][nt * 16 + lm];
                    b.y = sW[k + koff + 1][nt * 16 + lm];
                    c = __builtin_amdgcn_wmma_f32_16x16x4_f32(false, a, false, b,
                                                              (short)0, c, false, false);
                }
                acc[j] = c;
            }
        }
        __syncthreads();  // everyone finished reading sH

#pragma unroll
        for (int j = 0; j < 4; ++j) {
            int tile = wave + j * 8;
            if (tile < 28) {
                int mt = tile / 7, nt = tile % 7;
                int ncol  = nt * 16 + lm;
                int mbase = mt * 16 + (lane >> 4) * 8;
                if (ncol < H_) {
#pragma unroll
                    for (int r = 0; r < 8; ++r) {
                        int m = mbase + r;
                        size_t g = ((size_t)m * T_ + t) * H_ + ncol;
                        float v = tanhf(acc[j][r] + xp[g]);
                        sH[m][ncol] = v;
                        out[g] = v;
                    }
                }
            }
        }
        __syncthreads();  // h fully updated before next step
    }

    for (int i = tid; i < B_ * H_; i += 256) {
        int m = i / H_, n = i % H_;
        hT[i] = sH[m][n];
    }
}

// ---------------------------------------------------------------------------
// Output projection: y[m,o] = sum_h out[m,h] * Wout[o,h] + bout[o]   (N=10)
// ---------------------------------------------------------------------------
__global__ void proj_kernel(const float* __restrict__ outp, const float* __restrict__ Wout,
                            const float* __restrict__ bout, float* __restrict__ y) {
    int idx = blockIdx.x * blockDim.x + threadIdx.x;  // over M_*OUT_
    if (idx >= M_ * OUT_) return;
    int o = idx % OUT_;
    int m = idx / OUT_;
    const float* hr = outp + (size_t)m * H_;
    const float* wr = Wout + (size_t)o * H_;
    float acc = bout[o];
    for (int h = 0; h < H_; ++h) acc = fmaf(hr[h], wr[h], acc);
    y[idx] = acc;
}

// ---------------------------------------------------------------------------
extern "C" void kernel_launch(void* const* d_in, const int* in_sizes, int n_in,
                              void* d_out, int out_size, void* d_ws, size_t ws_size,
                              hipStream_t stream) {
    const float* x    = (const float*)d_in[0];  // [B,T,IN]
    const float* h0   = (const float*)d_in[1];  // [L,B,H]
    const float* Wih0 = (const float*)d_in[2];  // [H,IN]
    const float* Wih  = (const float*)d_in[3];  // [L-1,H,H]
    const float* Whh  = (const float*)d_in[4];  // [L,H,H]
    const float* bih  = (const float*)d_in[5];  // [L,H]
    const float* bhh  = (const float*)d_in[6];  // [L,H]
    const float* Wout = (const float*)d_in[7];  // [OUT,H]
    const float* bout = (const float*)d_in[8];  // [OUT]

    float* y      = (float*)d_out;              // [B,T,OUT]
    float* hstack = y + (size_t)M_ * OUT_;      // [L,B,H]

    float* xpA  = (float*)d_ws;                 // [B,T,H] input contributions
    float* outB = xpA + (size_t)M_ * H_;        // [B,T,H] layer output

    xp0_kernel<<<(M_ * H_ + 255) / 256, 256, 0, stream>>>(x, Wih0, bih, bhh, xpA);
    scan_kernel<<<1, 256, 0, stream>>>(xpA, outB, h0, Whh, hstack);

    for (int l = 1; l < L_; ++l) {
        xp_gemm_kernel<<<M_ / 16, 224, 0, stream>>>(outB, Wih + (size_t)(l - 1) * H_ * H_,
                                                    bih + l * H_, bhh + l * H_, xpA);
        scan_kernel<<<1, 256, 0, stream>>>(xpA, outB, h0 + (size_t)l * B_ * H_,
                                           Whh + (size_t)l * H_ * H_,
                                           hstack + (size_t)l * B_ * H_);
    }

    proj_kernel<<<(M_ * OUT_ + 255) / 256, 256, 0, stream>>>(outB, Wout, bout, y);
}